// Skeleton_4398046511108
// MI455X (gfx1250) — compile-verified
//
#include <hip/hip_runtime.h>

// Skeletonization: skel = sum_{k=0..20} [ e_k - dilate(e_{k+1}) ],  e_k = erode^k(x)
// erode = 3x3 min with +inf OOB padding per application; dilate = 3x3 max with -inf OOB.
// Fully fused: one workgroup per 64x64 output tile, 22-pixel halo staged in LDS,
// 21 in-LDS erode steps with shrinking valid radius, register accumulation.

#define IMG       1024
#define NBATCH    16
#define TILE      64
#define GRIDT     (IMG / TILE)          // 16
#define NITER     21                    // contributions k = 0..20
#define HALO      (NITER + 1)           // 22
#define SDIM      (TILE + 2 * HALO)     // 108
#define NTHREADS  256
#define PPT       ((TILE * TILE) / NTHREADS)  // 16 output pixels per thread

#if defined(__gfx1250__) && __has_builtin(__builtin_amdgcn_global_load_async_to_lds_b32)
#define USE_ASYNC_LDS 1
#else
#define USE_ASYNC_LDS 0
#endif

#if defined(__HIP_DEVICE_COMPILE__) && !USE_ASYNC_LDS
#warning "gfx1250 async global->LDS builtin unavailable; falling back to plain LDS stores"
#endif

#if USE_ASYNC_LDS
// Builtin signature (from hipcc diagnostic): (as1 int*, as3 int*, imm offset, imm cpol)
typedef __attribute__((address_space(1))) int gas1_int;
typedef __attribute__((address_space(3))) int las3_int;

__device__ __forceinline__ void async_copy_f32(const float* gp, float* lp) {
  // per-lane: copies 4 bytes from the global address in gp to the LDS address in lp, tracked by ASYNCcnt
  __builtin_amdgcn_global_load_async_to_lds_b32((gas1_int*)gp, (las3_int*)lp, 0, 0);
}

__device__ __forceinline__ void async_wait_all() {
#if __has_builtin(__builtin_amdgcn_s_wait_asynccnt)
  __builtin_amdgcn_s_wait_asynccnt(0);
#else
  asm volatile("s_wait_asynccnt 0" ::: "memory");
#endif
}
#endif

// One erosion step: B[r,c] = min 3x3 of A around (r,c) over the shrinking valid
// square (offset o, size R). For border tiles, out-of-image positions forced to +inf
// (matches per-application +inf padding of the reference erode).
template <bool BORDER>
__device__ __forceinline__ void erode_step(const float* __restrict__ A,
                                           float* __restrict__ B,
                                           int o, int R, int gr0, int gc0,
                                           int tidr, int tidc) {
  const float PINF = __builtin_inff();
  for (int r = o + tidr; r < o + R; r += 8) {
    const float* rm = A + (r - 1) * SDIM;
    const float* r0 = A + r * SDIM;
    const float* rp = A + (r + 1) * SDIM;
    for (int c = o + tidc; c < o + R; c += 32) {
      float m = fminf(fminf(rm[c - 1], rm[c]), rm[c + 1]);
      m = fminf(m, fminf(fminf(r0[c - 1], r0[c]), r0[c + 1]));
      m = fminf(m, fminf(fminf(rp[c - 1], rp[c]), rp[c + 1]));
      if (BORDER) {
        int gr = gr0 + r, gc = gc0 + c;
        bool inb = (gr >= 0) && (gr < IMG) && (gc >= 0) && (gc < IMG);
        m = inb ? m : PINF;
      }
      B[r * SDIM + c] = m;
    }
  }
}

// Accumulate acc[i] += A[p] - dilate(B)(p) over this thread's output pixels.
// dilate uses -inf for out-of-image neighbors (B holds +inf there, so must mask).
template <bool BORDER>
__device__ __forceinline__ void contrib_step(const float* __restrict__ A,
                                             const float* __restrict__ B,
                                             float* __restrict__ acc,
                                             int tid, int gy0, int gx0) {
  const float NINF = -__builtin_inff();
#pragma unroll
  for (int i = 0; i < PPT; ++i) {
    int pidx = tid + i * NTHREADS;
    int r = pidx >> 6;          // TILE == 64
    int c = pidx & (TILE - 1);
    int pr = HALO + r, pc = HALO + c;
    const float* bm = B + (pr - 1) * SDIM + pc;
    const float* b0 = B + pr * SDIM + pc;
    const float* bp = B + (pr + 1) * SDIM + pc;
    float d;
    if (!BORDER) {
      d = fmaxf(fmaxf(bm[-1], bm[0]), bm[1]);
      d = fmaxf(d, fmaxf(fmaxf(b0[-1], b0[0]), b0[1]));
      d = fmaxf(d, fmaxf(fmaxf(bp[-1], bp[0]), bp[1]));
    } else {
      int gr = gy0 + r, gc = gx0 + c;      // always in-image (output pixel)
      bool rmv = (gr - 1 >= 0), rpv = (gr + 1 < IMG);
      bool cmv = (gc - 1 >= 0), cpv = (gc + 1 < IMG);
      d = b0[0];
      d = fmaxf(d, cmv ? b0[-1] : NINF);
      d = fmaxf(d, cpv ? b0[1] : NINF);
      d = fmaxf(d, rmv ? bm[0] : NINF);
      d = fmaxf(d, (rmv && cmv) ? bm[-1] : NINF);
      d = fmaxf(d, (rmv && cpv) ? bm[1] : NINF);
      d = fmaxf(d, rpv ? bp[0] : NINF);
      d = fmaxf(d, (rpv && cmv) ? bp[-1] : NINF);
      d = fmaxf(d, (rpv && cpv) ? bp[1] : NINF);
    }
    acc[i] += A[pr * SDIM + pc] - d;
  }
}

__global__ __launch_bounds__(NTHREADS) void skeleton_kernel(
    const float* __restrict__ x, float* __restrict__ out) {
  __shared__ float bufA[SDIM * SDIM];
  __shared__ float bufB[SDIM * SDIM];

  const int tid = threadIdx.x;
  const int tx = blockIdx.x, ty = blockIdx.y, b = blockIdx.z;
  const bool border = (tx == 0) || (ty == 0) || (tx == GRIDT - 1) || (ty == GRIDT - 1);
  const int gr0 = ty * TILE - HALO;           // global row of LDS row 0
  const int gc0 = tx * TILE - HALO;           // global col of LDS col 0
  const float PINF = __builtin_inff();
  const size_t plane = (size_t)IMG * IMG;
  const float* src = x + (size_t)b * plane;

  // ---- Stage input tile (+halo) into LDS; out-of-image -> +inf ----
  if (!border) {
    for (int idx = tid; idx < SDIM * SDIM; idx += NTHREADS) {
      int r = idx / SDIM;
      int c = idx - r * SDIM;
      const float* gp = src + (size_t)(gr0 + r) * IMG + (gc0 + c);
#if USE_ASYNC_LDS
      async_copy_f32(gp, &bufA[idx]);
#else
      bufA[idx] = *gp;
#endif
    }
  } else {
    for (int idx = tid; idx < SDIM * SDIM; idx += NTHREADS) {
      int r = idx / SDIM;
      int c = idx - r * SDIM;
      int gr = gr0 + r, gc = gc0 + c;
      bool inb = (gr >= 0) && (gr < IMG) && (gc >= 0) && (gc < IMG);
      if (inb) {
        const float* gp = src + (size_t)gr * IMG + gc;
#if USE_ASYNC_LDS
        async_copy_f32(gp, &bufA[idx]);
#else
        bufA[idx] = *gp;
#endif
      } else {
        bufA[idx] = PINF;
      }
    }
  }
#if USE_ASYNC_LDS
  async_wait_all();  // each wave drains its ASYNCcnt before signaling the barrier
#endif
  __syncthreads();

  // ---- 21 fused (erode, dilate-and-accumulate) steps ----
  float acc[PPT];
#pragma unroll
  for (int i = 0; i < PPT; ++i) acc[i] = 0.0f;

  float* A = bufA;  // holds e_j
  float* B = bufB;  // receives e_{j+1}
  const int tidr = tid >> 5;        // 8 row groups
  const int tidc = tid & 31;        // 32 cols (one wave32 = one row slice)
  const int gy0 = ty * TILE, gx0 = tx * TILE;

  for (int j = 0; j < NITER; ++j) {
    const int n = NITER - j;        // valid radius of e_{j+1}: 21 .. 1
    const int o = HALO - n;
    const int R = TILE + 2 * n;

    if (border) erode_step<true>(A, B, o, R, gr0, gc0, tidr, tidc);
    else        erode_step<false>(A, B, o, R, gr0, gc0, tidr, tidc);
    __syncthreads();

    if (border) contrib_step<true>(A, B, acc, tid, gy0, gx0);
    else        contrib_step<false>(A, B, acc, tid, gy0, gx0);
    __syncthreads();

    float* t = A; A = B; B = t;
  }

  // ---- Store skel tile ----
  float* dst = out + (size_t)b * plane;
#pragma unroll
  for (int i = 0; i < PPT; ++i) {
    int pidx = tid + i * NTHREADS;
    int r = pidx >> 6;
    int c = pidx & (TILE - 1);
    dst[(size_t)(gy0 + r) * IMG + (gx0 + c)] = acc[i];
  }
}

extern "C" void kernel_launch(void* const* d_in, const int* in_sizes, int n_in,
                              void* d_out, int out_size, void* d_ws, size_t ws_size,
                              hipStream_t stream) {
  const float* x = (const float*)d_in[0];
  float* out = (float*)d_out;
  (void)in_sizes; (void)n_in; (void)out_size; (void)d_ws; (void)ws_size;
  dim3 grid(GRIDT, GRIDT, NBATCH);
  skeleton_kernel<<<grid, dim3(NTHREADS, 1, 1), 0, stream>>>(x, out);
}